// SeqModel_51230369907301
// MI455X (gfx1250) — compile-verified
//
#include <hip/hip_runtime.h>

// Problem constants (match reference)
#define B_   64
#define T_   4096
#define F_   100
#define H_   200
#define G3_  600     // 3*H
#define NWG  13      // ceil(H/16) column tiles, one WG each
#define TPB  128     // 4 waves: one 16-row tile of B each

typedef float v2f __attribute__((ext_vector_type(2)));
typedef float v8f __attribute__((ext_vector_type(8)));

// D = A(16x4 f32) * B(4x16 f32) + C(16x16 f32)
__device__ inline v8f wmma_f32(v2f a, v2f b, v8f c) {
    return __builtin_amdgcn_wmma_f32_16x16x4_f32(
        /*neg_a=*/false, a, /*neg_b=*/false, b,
        /*c_mod=*/(short)0, c, /*reuse_a=*/false, /*reuse_b=*/false);
}

// Generation-counter grid barrier across NWG co-resident workgroups.
// bar[0] = arrival counter, bar[1] = generation.
__device__ inline void grid_sync(unsigned* bar) {
    __syncthreads();
    if (threadIdx.x == 0) {
        unsigned gen  = __hip_atomic_load(&bar[1], __ATOMIC_ACQUIRE, __HIP_MEMORY_SCOPE_AGENT);
        unsigned prev = __hip_atomic_fetch_add(&bar[0], 1u, __ATOMIC_ACQ_REL, __HIP_MEMORY_SCOPE_AGENT);
        if (prev == (unsigned)(NWG - 1)) {
            __hip_atomic_store(&bar[0], 0u, __ATOMIC_RELAXED, __HIP_MEMORY_SCOPE_AGENT);
            __hip_atomic_store(&bar[1], gen + 1u, __ATOMIC_RELEASE, __HIP_MEMORY_SCOPE_AGENT);
        } else {
            while (__hip_atomic_load(&bar[1], __ATOMIC_ACQUIRE, __HIP_MEMORY_SCOPE_AGENT) == gen) {
                __builtin_amdgcn_s_sleep(2);
            }
        }
    }
    __syncthreads();
}

// out[b*T+t] = bo[0]; zero barrier + h double-buffer region of workspace.
__global__ void gru_init_kernel(float* __restrict__ out, const float* __restrict__ bo,
                                float* __restrict__ wsf, int nws) {
    int i = blockIdx.x * blockDim.x + threadIdx.x;
    if (i < B_ * T_) out[i] = bo[0];
    if (i < nws)     wsf[i] = 0.0f;
}

// Persistent fused GRU: 13 WGs x 4 waves; WG owns a 16-col tile of H, wave owns
// a 16-row tile of B. Wi/Wh column slices live in LDS (col-major) for the whole
// kernel; h ping-pongs in global workspace with one grid barrier per timestep.
__global__ __launch_bounds__(TPB) void gru_persistent_kernel(
    const float* __restrict__ x,   // [B, T, F]
    const float* __restrict__ Wi,  // [F, 3H]
    const float* __restrict__ bi,  // [3H]
    const float* __restrict__ Wh,  // [H, 3H]
    const float* __restrict__ bhn, // [H]
    const float* __restrict__ Wo,  // [H, 1]
    float* __restrict__ out,       // [B, T] (O==1)
    float* __restrict__ hbuf,      // 2 * [B, H] double buffer
    unsigned* __restrict__ bar)    // 2 unsigned
{
    // LDS: column-major weight slices. c = gate*16 + n  (48 cols), zero-padded for j>=H.
    __shared__ float WiL[48 * F_];   // 19.2 KB
    __shared__ float WhL[48 * H_];   // 38.4 KB

    const int wg   = blockIdx.x;
    const int j0   = wg * 16;          // column tile base within H
    const int lane = threadIdx.x & 31;
    const int wave = threadIdx.x >> 5;
    const int m0   = wave * 16;        // row (batch) tile base
    const int hi   = lane >> 4;        // half-wave selector
    const int n    = lane & 15;        // column within tile
    const int j    = j0 + n;           // global hidden column
    const bool jv  = (j < H_);

    // ---- stage weight slices into LDS (col-major: [c][K]) ----
    for (int idx = threadIdx.x; idx < 48 * F_; idx += TPB) {
        int c = idx / F_, k = idx - c * F_;
        int g = c >> 4, cn = c & 15, jj = j0 + cn;
        WiL[idx] = (jj < H_) ? Wi[(size_t)k * G3_ + g * H_ + jj] : 0.0f;
    }
    for (int idx = threadIdx.x; idx < 48 * H_; idx += TPB) {
        int c = idx / H_, k = idx - c * H_;
        int g = c >> 4, cn = c & 15, jj = j0 + cn;
        WhL[idx] = (jj < H_) ? Wh[(size_t)k * G3_ + g * H_ + jj] : 0.0f;
    }
    __syncthreads();

    // Per-lane loop-invariant scalars for this hidden column.
    const float bir   = jv ? bi[j]           : 0.0f;
    const float biz   = jv ? bi[H_ + j]      : 0.0f;
    const float bin   = jv ? bi[2 * H_ + j]  : 0.0f;
    const float bhn_l = jv ? bhn[j]          : 0.0f;
    const float wo_l  = jv ? Wo[j]           : 0.0f;

    const int bA = m0 + n;                       // A-fragment row this lane feeds
    const float* xA = x + (size_t)bA * T_ * F_;  // row base in x

    for (int t = 0; t < T_; ++t) {
        float* hcur = hbuf + (size_t)(t & 1) * (B_ * H_);
        float* hnew = hbuf + (size_t)((t + 1) & 1) * (B_ * H_);

        v8f ar = {}, az = {}, agn = {}, ahn = {};   // r/z fused (gi+hh); n split

        // ---- gi = x[:,t,:] @ Wi  (K = F = 100, 25 k-steps) ----
        const float* xrow = xA + (size_t)t * F_;
        if (t + 1 < T_) __builtin_prefetch(xrow + F_, 0, 1);  // next-step x row
        #pragma unroll 5
        for (int ks = 0; ks < F_ / 4; ++ks) {
            int k = 4 * ks + 2 * hi;
            v2f a  = *(const v2f*)(xrow + k);
            v2f br = *(const v2f*)&WiL[(0  + n) * F_ + k];
            v2f bz = *(const v2f*)&WiL[(16 + n) * F_ + k];
            v2f bn = *(const v2f*)&WiL[(32 + n) * F_ + k];
            ar  = wmma_f32(a, br, ar);
            az  = wmma_f32(a, bz, az);
            agn = wmma_f32(a, bn, agn);
        }

        // ---- hh = h @ Wh  (K = H = 200, 50 k-steps) ----
        const float* hrow = hcur + (size_t)bA * H_;
        #pragma unroll 5
        for (int ks = 0; ks < H_ / 4; ++ks) {
            int k = 4 * ks + 2 * hi;
            v2f a  = *(const v2f*)(hrow + k);
            v2f br = *(const v2f*)&WhL[(0  + n) * H_ + k];
            v2f bz = *(const v2f*)&WhL[(16 + n) * H_ + k];
            v2f bn = *(const v2f*)&WhL[(32 + n) * H_ + k];
            ar  = wmma_f32(a, br, ar);
            az  = wmma_f32(a, bz, az);
            ahn = wmma_f32(a, bn, ahn);
        }

        // ---- gates + h update + fused output dot (C/D row = i + 8*hi) ----
        #pragma unroll
        for (int i = 0; i < 8; ++i) {
            int brow = m0 + i + 8 * hi;
            float r    = 1.0f / (1.0f + __expf(-(ar[i] + bir)));
            float z    = 1.0f / (1.0f + __expf(-(az[i] + biz)));
            float cand = tanhf(agn[i] + bin + r * (ahn[i] + bhn_l));
            float hp   = jv ? hcur[(size_t)brow * H_ + j] : 0.0f;
            float hv   = (1.0f - z) * cand + z * hp;
            if (jv) hnew[(size_t)brow * H_ + j] = hv;

            // out[b,t] += sum_j hv * Wo[j] over this 16-col tile
            float oc = jv ? hv * wo_l : 0.0f;
            oc += __shfl_xor(oc, 1, 32);
            oc += __shfl_xor(oc, 2, 32);
            oc += __shfl_xor(oc, 4, 32);
            oc += __shfl_xor(oc, 8, 32);
            if (n == 0) atomicAdd(out + (size_t)brow * T_ + t, oc);
        }

        __threadfence();   // make hnew/out globally visible before arriving
        grid_sync(bar);
    }
}

extern "C" void kernel_launch(void* const* d_in, const int* in_sizes, int n_in,
                              void* d_out, int out_size, void* d_ws, size_t ws_size,
                              hipStream_t stream) {
    const float* x   = (const float*)d_in[0];
    const float* Wi  = (const float*)d_in[1];
    const float* bi  = (const float*)d_in[2];
    const float* Wh  = (const float*)d_in[3];
    const float* bhn = (const float*)d_in[4];
    const float* Wo  = (const float*)d_in[5];
    const float* bo  = (const float*)d_in[6];
    float* out = (float*)d_out;

    // Workspace layout: [0,8)   barrier (2 x u32), padded to 256 B
    //                   [256,..) h double buffer: 2 * B*H floats (100 KB)
    unsigned* bar = (unsigned*)d_ws;
    float* hbuf = (float*)((char*)d_ws + 256);
    const int nws = 64 + 2 * B_ * H_;   // floats to zero (barrier region + h buffers)

    gru_init_kernel<<<(B_ * T_ + 255) / 256, 256, 0, stream>>>(out, bo, (float*)d_ws, nws);
    gru_persistent_kernel<<<NWG, TPB, 0, stream>>>(x, Wi, bi, Wh, bhn, Wo, out, hbuf, bar);
}